// Net_27358941676189
// MI455X (gfx1250) — compile-verified
//
#include <hip/hip_runtime.h>
#include <hip/hip_bf16.h>
#include <math.h>

#define GN 2048
#define EL 512
#define BN 16

typedef float v2f __attribute__((ext_vector_type(2)));
typedef float v8f __attribute__((ext_vector_type(8)));

__device__ __forceinline__ float lrelu(float x) { return x > 0.f ? x : 0.01f * x; }

// ---------------- K0: pack xbin = (datax != 0) into bitmasks ----------------
__global__ void k_bits(const int* __restrict__ datax, unsigned* __restrict__ bits) {
    int w = blockIdx.x * blockDim.x + threadIdx.x;      // word index, B*(G/32) = 1024
    if (w >= BN * (GN / 32)) return;
    int b = w >> 6, wi = w & 63;
    const int* p = datax + b * GN + wi * 32;
    unsigned m = 0u;
#pragma unroll
    for (int j = 0; j < 32; ++j) m |= (p[j] != 0 ? 1u : 0u) << j;
    bits[w] = m;
}

// ---------------- K1: q/k projections via v_wmma_f32_16x16x4_f32, then s[2][2]
// A tile (16x4): rows 0/1 = embMat rows 0/1, rows 2..15 zero (mask-multiplied,
// loads kept unconditional so EXEC stays all-ones around the WMMAs).
// B tile (4x16): B[k][n] = W[f=nt*16+n][kc+k]   (Linear: q[f] = sum_e x[e]*W[f][e])
__global__ void k_proj(const float* __restrict__ embMat,
                       const float* __restrict__ Wq,
                       const float* __restrict__ Wk,
                       float* __restrict__ s_out) {
    __shared__ float qs[2][EL];
    __shared__ float ks[2][EL];
    const int lane = threadIdx.x;          // single wave32
    const int m  = lane & 15;              // A row / B col (per half-wave)
    const int kb = (lane >> 4) * 2;        // K base: lanes0-15 -> 0, lanes16-31 -> 2
    const float amask = (m < 2) ? 1.f : 0.f;
    const float* arow = embMat + (size_t)(m < 2 ? m : 0) * EL;

    for (int nt = 0; nt < EL / 16; ++nt) {
        v8f accq = {};
        v8f acck = {};
        const int f = nt * 16 + m;
        for (int kc = 0; kc < EL; kc += 4) {
            v2f a, bq, bk;
            a.x = arow[kc + kb]     * amask;
            a.y = arow[kc + kb + 1] * amask;
            bq.x = Wq[(size_t)f * EL + kc + kb];
            bq.y = Wq[(size_t)f * EL + kc + kb + 1];
            bk.x = Wk[(size_t)f * EL + kc + kb];
            bk.y = Wk[(size_t)f * EL + kc + kb + 1];
            accq = __builtin_amdgcn_wmma_f32_16x16x4_f32(false, a, false, bq,
                                                         (short)0, accq, false, false);
            acck = __builtin_amdgcn_wmma_f32_16x16x4_f32(false, a, false, bk,
                                                         (short)0, acck, false, false);
        }
        // D layout: lanes 0-15 hold rows M=0..7 in regs 0..7; we need M=0 and M=1.
        if (lane < 16) {
            qs[0][nt * 16 + lane] = accq[0];
            qs[1][nt * 16 + lane] = accq[1];
            ks[0][nt * 16 + lane] = acck[0];
            ks[1][nt * 16 + lane] = acck[1];
        }
    }
    __syncthreads();

    // s[a][b] = dot(q_a, k_b) over EL, wave32 butterfly reduce
    float p00 = 0.f, p01 = 0.f, p10 = 0.f, p11 = 0.f;
    for (int fj = lane; fj < EL; fj += 32) {
        float q0 = qs[0][fj], q1 = qs[1][fj];
        float k0 = ks[0][fj], k1 = ks[1][fj];
        p00 += q0 * k0; p01 += q0 * k1;
        p10 += q1 * k0; p11 += q1 * k1;
    }
    for (int off = 16; off; off >>= 1) {
        p00 += __shfl_xor(p00, off, 32);
        p01 += __shfl_xor(p01, off, 32);
        p10 += __shfl_xor(p10, off, 32);
        p11 += __shfl_xor(p11, off, 32);
    }
    if (lane == 0) {
        s_out[0] = p00; s_out[1] = p01; s_out[2] = p10; s_out[3] = p11;
    }
}

// ------- K2: partial w_eff: part[p][g] = sum_{j in chunk p} fcCox[j]*fc1_w[j][g]
__global__ void k_weff_part(const float* __restrict__ fc1_w,
                            const float* __restrict__ fcCox,
                            float* __restrict__ part) {
    const int g = blockIdx.x * blockDim.x + threadIdx.x;   // grid.x = GN/256
    const int p = blockIdx.y;                              // 16 chunks of 128 rows
    float acc = 0.f;
    const int j0 = p * (GN / 16);
    for (int j = j0; j < j0 + GN / 16; ++j)
        acc += fcCox[j] * fc1_w[(size_t)j * GN + g];
    part[p * GN + g] = acc;
}

// ---------------- K3: fused bias + attn_origin store + softmax (factored) ----
// One block per g row; thread t owns h = t*8 .. t*8+7 (b128 loads/stores).
// score(b,h) = bias[h] + s[a_b][bit(b,h)].  softmax is shift-invariant, so with
// eb[h] = exp(bias[h]-bmax), E0/E1 = exp(s[a][.]-smax4):
//   den = E0*(Sall-S1_b) + E1*S1_b,  num = E1*S1_b,  S1_b = sum_h eb[h]*bit(b,h)
__global__ void __launch_bounds__(256) k_main(
        const float* __restrict__ k1m, const float* __restrict__ k2m,
        const float* __restrict__ k3v, const float* __restrict__ sp,
        const float* __restrict__ cm,  const float* __restrict__ pad,
        const unsigned* __restrict__ bits, const float* __restrict__ s4,
        float* __restrict__ outAttn, float* __restrict__ attnv) {
    __shared__ unsigned bl[BN * (GN / 32)];   // 1024 words = 4KB
    __shared__ float redm[8];                 // per-wave bias max
    __shared__ float reda[8];                 // per-wave eb sums
    __shared__ float redw[BN][8];             // per-(b, wave) S1 partials
    const int t = threadIdx.x;
    const int g = blockIdx.x;
    const int wid = t >> 5, lane = t & 31;

    for (int i = t; i < BN * (GN / 32); i += 256) bl[i] = bits[i];

    const float s00 = s4[0], s01 = s4[1], s10 = s4[2], s11 = s4[3];
    const float smax4 = fmaxf(fmaxf(s00, s01), fmaxf(s10, s11));
    const float e00 = __expf(s00 - smax4), e01 = __expf(s01 - smax4);
    const float e10 = __expf(s10 - smax4), e11 = __expf(s11 - smax4);
    const float b3 = lrelu(k3v[0]);

    // --- bias row, 8 contiguous floats per thread, b128 traffic ---
    const size_t rowb = (size_t)g * GN + (size_t)t * 8;
    const float4* k1p = (const float4*)(k1m + rowb);
    const float4* k2p = (const float4*)(k2m + rowb);
    const float4* spp = (const float4*)(sp  + rowb);
    const float4* cmp = (const float4*)(cm  + rowb);
    const float4* pdp = (const float4*)(pad + rowb);
    float bias_r[8];
#pragma unroll
    for (int q = 0; q < 2; ++q) {
        float4 a1 = k1p[q], a2 = k2p[q], as = spp[q], ac = cmp[q], ap = pdp[q];
        bias_r[q*4+0] = lrelu(a1.x)*as.x + lrelu(a2.x)*ac.x + b3*ap.x;
        bias_r[q*4+1] = lrelu(a1.y)*as.y + lrelu(a2.y)*ac.y + b3*ap.y;
        bias_r[q*4+2] = lrelu(a1.z)*as.z + lrelu(a2.z)*ac.z + b3*ap.z;
        bias_r[q*4+3] = lrelu(a1.w)*as.w + lrelu(a2.w)*ac.w + b3*ap.w;
    }

    // --- row max of bias (wave shuffle + 8-entry LDS combine) ---
    float lmax = bias_r[0];
#pragma unroll
    for (int i = 1; i < 8; ++i) lmax = fmaxf(lmax, bias_r[i]);
    for (int off = 16; off; off >>= 1) lmax = fmaxf(lmax, __shfl_xor(lmax, off, 32));
    if (lane == 0) redm[wid] = lmax;
    __syncthreads();                      // also publishes bl[]
    float bmax = redm[0];
#pragma unroll
    for (int w = 1; w < 8; ++w) bmax = fmaxf(bmax, redm[w]);

    // --- eb and its row sum ---
    float eb[8]; float ebsum = 0.f;
#pragma unroll
    for (int i = 0; i < 8; ++i) { eb[i] = __expf(bias_r[i] - bmax); ebsum += eb[i]; }
    for (int off = 16; off; off >>= 1) ebsum += __shfl_xor(ebsum, off, 32);
    if (lane == 0) reda[wid] = ebsum;

    // --- per-batch: emit score plane (b128 stores) + accumulate S1 partial ---
    const int word = t >> 2;              // h/32 for h = t*8
    const int shift = (t & 3) * 8;
    float s1p[BN];
#pragma unroll
    for (int b = 0; b < BN; ++b) {
        const unsigned hbyte = (bl[b * 64 + word] >> shift) & 0xFFu;
        const int a = (bl[b * 64 + (g >> 5)] >> (g & 31)) & 1;
        const float sa0 = a ? s10 : s00;
        const float sa1 = a ? s11 : s01;
        float vv[8]; float p = 0.f;
#pragma unroll
        for (int i = 0; i < 8; ++i) {
            const int vb = (hbyte >> i) & 1;
            vv[i] = bias_r[i] + (vb ? sa1 : sa0);
            p += vb ? eb[i] : 0.f;
        }
        float4* op = (float4*)(outAttn + ((size_t)b * GN + g) * GN + (size_t)t * 8);
        float4 o0 = {vv[0], vv[1], vv[2], vv[3]};
        float4 o1 = {vv[4], vv[5], vv[6], vv[7]};
        op[0] = o0; op[1] = o1;
        s1p[b] = p;
    }
#pragma unroll
    for (int b = 0; b < BN; ++b)
        for (int off = 16; off; off >>= 1) s1p[b] += __shfl_xor(s1p[b], off, 32);
    if (lane == 0) {
#pragma unroll
        for (int b = 0; b < BN; ++b) redw[b][wid] = s1p[b];
    }
    __syncthreads();

    // --- threads 0..15 finalize their batch ---
    if (t < BN) {
        float Sall = 0.f, S1 = 0.f;
#pragma unroll
        for (int w = 0; w < 8; ++w) { Sall += reda[w]; S1 += redw[t][w]; }
        const int a = (bl[t * 64 + (g >> 5)] >> (g & 31)) & 1;
        const float E0 = a ? e10 : e00;
        const float E1 = a ? e11 : e01;
        const float den = E0 * (Sall - S1) + E1 * S1;
        const float num = E1 * S1;
        attnv[t * GN + g] = (float)a + num / den;   // originalx + attn
    }
}

// ---------------- K4: output[b] = c0 + dot(attnv[b,:], w_eff) ----------------
__global__ void k_final(const float* __restrict__ fcCox,
                        const float* __restrict__ fc1_b,
                        const float* __restrict__ attnv,
                        const float* __restrict__ part,
                        float* __restrict__ out) {
    __shared__ float weffs[GN];          // 8KB
    __shared__ float red[256];
    const int t = threadIdx.x;
    for (int g = t; g < GN; g += 256) {
        float s = 0.f;
#pragma unroll
        for (int p = 0; p < 16; ++p) s += part[p * GN + g];
        weffs[g] = s;
    }
    float c = 0.f;
    for (int j = t; j < GN; j += 256) c += fcCox[j] * fc1_b[j];
    red[t] = c; __syncthreads();         // also publishes weffs
    for (int off = 128; off; off >>= 1) {
        if (t < off) red[t] += red[t + off];
        __syncthreads();
    }
    const float c0 = red[0]; __syncthreads();

    for (int b = 0; b < BN; ++b) {
        float p = 0.f;
        for (int g = t; g < GN; g += 256) p += attnv[b * GN + g] * weffs[g];
        red[t] = p; __syncthreads();
        for (int off = 128; off; off >>= 1) {
            if (t < off) red[t] += red[t + off];
            __syncthreads();
        }
        if (t == 0) out[b] = c0 + red[0];
        __syncthreads();
    }
}

extern "C" void kernel_launch(void* const* d_in, const int* in_sizes, int n_in,
                              void* d_out, int out_size, void* d_ws, size_t ws_size,
                              hipStream_t stream) {
    (void)in_sizes; (void)n_in; (void)out_size; (void)ws_size;
    const int*   datax  = (const int*)  d_in[0];
    const float* embMat = (const float*)d_in[1];
    const float* Wq     = (const float*)d_in[2];
    const float* Wk     = (const float*)d_in[3];
    const float* k1m    = (const float*)d_in[4];
    const float* k2m    = (const float*)d_in[5];
    const float* k3v    = (const float*)d_in[6];
    const float* sp     = (const float*)d_in[7];
    const float* cm     = (const float*)d_in[8];
    const float* pad    = (const float*)d_in[9];
    const float* fc1_w  = (const float*)d_in[10];
    const float* fc1_b  = (const float*)d_in[11];
    const float* fcCox  = (const float*)d_in[12];
    float* out = (float*)d_out;

    char* ws = (char*)d_ws;
    float*    s4    = (float*)(ws);                          // 4 floats
    unsigned* bits  = (unsigned*)(ws + 256);                 // 1024 u32 = 4KB
    float*    attnv = (float*)(ws + 256 + 4096);             // B*G floats = 128KB
    float*    part  = (float*)(ws + 256 + 4096
                               + (size_t)BN * GN * sizeof(float)); // 16*G = 128KB

    k_bits     <<<4, 256, 0, stream>>>(datax, bits);
    k_proj     <<<1, 32, 0, stream>>>(embMat, Wq, Wk, s4);
    k_weff_part<<<dim3(GN / 256, 16), 256, 0, stream>>>(fc1_w, fcCox, part);
    k_main     <<<GN, 256, 0, stream>>>(k1m, k2m, k3v, sp, cm, pad, bits, s4,
                                        out + BN, attnv);
    k_final    <<<1, 256, 0, stream>>>(fcCox, fc1_b, attnv, part, out);
}